// DocREModel_50190987821609
// MI455X (gfx1250) — compile-verified
//
#include <hip/hip_runtime.h>
#include <hip/hip_bf16.h>
#include <math.h>

// ---------------- problem constants ----------------
constexpr int Bc = 4, Lc = 1024, Dc = 768, Hc = 12;
constexpr int Ec = 24, Mc = 4, Pc = 552;
constexpr int EMBc = 768, BLKc = 64, NLc = 97;
constexpr int Rc = Bc * Pc;          // 2208 (= 138 * 16 exactly)
constexpr int KBLK = EMBc / BLKc;    // 12

// ---------------- workspace layout (floats), lifetime-aliased ----------------
constexpr size_t SZ_ENT = (size_t)Bc * Ec * Hc * Lc;   // 1,179,648
constexpr size_t SZ_HT  = (size_t)Bc * Pc * Lc;        // 2,260,992
constexpr size_t SZ_RD  = (size_t)Rc * Dc;             // 1,695,744
constexpr size_t OFF_ENT = 0;
constexpr size_t OFF_HT  = OFF_ENT + SZ_ENT;
constexpr size_t OFF_RS  = OFF_HT  + SZ_HT;
constexpr size_t OFF_QC  = OFF_RS  + SZ_RD;
constexpr size_t OFF_QK  = OFF_QC  + SZ_RD;
constexpr size_t OFF_HE  = OFF_QK  + SZ_RD;
constexpr size_t OFF_TE  = OFF_HE  + SZ_RD;
constexpr size_t OFF_HS  = OFF_HT;   // ht_att dead after rs GEMM
constexpr size_t OFF_TS  = OFF_QC;   // q_c dead after qk GEMM
// total: ~47.7 MB

typedef __attribute__((ext_vector_type(2))) float v2f;
typedef __attribute__((ext_vector_type(8))) float v8f;

#define WMMA_F32(a, b, c) \
  __builtin_amdgcn_wmma_f32_16x16x4_f32(false, (a), false, (b), (short)0, (c), false, false)

// ---------------- K1: ent_att[b,e,h,l] = mean_m attention[b,h,pos(b,e,m),l] ----------------
__global__ void k_ent(const float* __restrict__ att, const int* __restrict__ mpos,
                      float* __restrict__ ent) {
  int idx = blockIdx.x;            // (b*E + e)*H + h
  int h  = idx % Hc;
  int be = idx / Hc;               // b*E + e
  int b  = be / Ec;
  int p0 = mpos[be * Mc + 0] + 1;
  int p1 = mpos[be * Mc + 1] + 1;
  int p2 = mpos[be * Mc + 2] + 1;
  int p3 = mpos[be * Mc + 3] + 1;
  const float* base = att + (size_t)(b * Hc + h) * Lc * Lc;
  float* dst = ent + ((size_t)be * Hc + h) * Lc;
  for (int l = threadIdx.x; l < Lc; l += blockDim.x) {
    float s = base[(size_t)p0 * Lc + l] + base[(size_t)p1 * Lc + l] +
              base[(size_t)p2 * Lc + l] + base[(size_t)p3 * Lc + l];
    dst[l] = s * 0.25f;
  }
}

// ---------------- K2: ht_att[b,p,l] = norm_l( mean_h h_att*t_att ) ----------------
__global__ void k_ht(const float* __restrict__ ent, const int* __restrict__ hts,
                     float* __restrict__ ht) {
  int r = blockIdx.x;              // b*P + p
  int b = r / Pc;
  int hi = hts[r * 2 + 0];
  int ti = hts[r * 2 + 1];
  const float* ha = ent + (size_t)(b * Ec + hi) * Hc * Lc;
  const float* ta = ent + (size_t)(b * Ec + ti) * Hc * Lc;
  __shared__ float red[256];
  float vals[4];
  float tot = 0.f;
  #pragma unroll
  for (int cch = 0; cch < 4; ++cch) {
    int l = threadIdx.x + cch * 256;
    float acc = 0.f;
    #pragma unroll
    for (int h = 0; h < Hc; ++h) acc += ha[(size_t)h * Lc + l] * ta[(size_t)h * Lc + l];
    acc *= (1.0f / Hc);
    vals[cch] = acc;
    tot += acc;
  }
  red[threadIdx.x] = tot;
  __syncthreads();
  for (int s = 128; s > 0; s >>= 1) {
    if (threadIdx.x < s) red[threadIdx.x] += red[threadIdx.x + s];
    __syncthreads();
  }
  float inv = 1.0f / (red[0] + 1e-5f);
  #pragma unroll
  for (int cch = 0; cch < 4; ++cch) {
    int l = threadIdx.x + cch * 256;
    ht[(size_t)r * Lc + l] = vals[cch] * inv;
  }
}

// ---------------- K3: rs[b] = ht_att[b] (P x L) @ seq[b] (L x D), fp32 WMMA ----------------
// Out-of-range A rows are CLAMPED (not zeroed): their D rows are garbage but never stored,
// and WMMA rows are independent -> no EXEC divergence in the k-loop.
__global__ void k_rs(const float* __restrict__ ht, const float* __restrict__ seq,
                     float* __restrict__ rs) {
  constexpr int MT = (Pc + 15) / 16;  // 35
  constexpr int NT = Dc / 16;         // 48
  int wid = blockIdx.x * (blockDim.x >> 5) + (threadIdx.x >> 5);
  if (wid >= Bc * MT * NT) return;    // wave-uniform
  int b   = wid / (MT * NT);
  int rem = wid - b * MT * NT;
  int mt = rem / NT, nt = rem - mt * NT;
  int lane = threadIdx.x & 31;
  int nl = lane & 15, kh = lane >> 4;
  int row = mt * 16 + nl;
  int rowc = row < Pc ? row : (Pc - 1);   // clamp, no divergence in loop
  int col = nt * 16 + nl;
  const float* ar = ht  + (size_t)b * Pc * Lc + (size_t)rowc * Lc + 2 * kh;
  const float* br = seq + (size_t)b * Lc * Dc + (size_t)(2 * kh) * Dc + col;
  v8f c = {};
  #pragma unroll 4
  for (int k = 0; k < Lc; k += 4) {
    v2f a, bf;
    a.x  = ar[k];
    a.y  = ar[k + 1];
    bf.x = br[(size_t)k * Dc];
    bf.y = br[(size_t)(k + 1) * Dc];
    c = WMMA_F32(a, bf, c);
  }
  float* rb = rs + ((size_t)b * Pc + (size_t)mt * 16 + 8 * kh) * Dc + col;
  if (mt != MT - 1) {                 // wave-uniform: full tile, no divergence
    #pragma unroll
    for (int v = 0; v < 8; ++v) rb[(size_t)v * Dc] = c[v];
  } else {                            // last M tile: rows 544..559, P=552
    #pragma unroll
    for (int v = 0; v < 8; ++v) {
      int orow = mt * 16 + v + 8 * kh;
      if (orow < Pc) rb[(size_t)v * Dc] = c[v];
    }
  }
}

// ---------------- K4/K6: generic fp32 WMMA GEMM, A = [A1 | A2] split at K1 (two clean loops),
//                  optional B^T, optional bias + tanh epilogue.
//                  REQUIRES Rr % 16 == 0 (true for all call sites: R = 2208). ----------------
__global__ void k_gemm2(const float* __restrict__ A1, const float* __restrict__ A2, int K1,
                        const float* __restrict__ Bm, int bT,
                        const float* __restrict__ bias, int doTanh,
                        float* __restrict__ Cm, int Rr, int Kk, int Nn) {
  int MT = Rr >> 4, NT = Nn >> 4;
  int wid = blockIdx.x * (blockDim.x >> 5) + (threadIdx.x >> 5);
  if (wid >= MT * NT) return;        // wave-uniform
  int mt = wid / NT, nt = wid - mt * NT;
  int lane = threadIdx.x & 31;
  int nl = lane & 15, kh = lane >> 4;
  int row = mt * 16 + nl;
  int col = nt * 16 + nl;
  int K2s = Kk - K1;
  v8f c = {};
  if (bT) {
    const float* a1 = A1 + (size_t)row * K1 + 2 * kh;
    const float* bp = Bm + (size_t)col * Kk + 2 * kh;
    #pragma unroll 4
    for (int k = 0; k < K1; k += 4) {
      v2f a, b;
      a.x = a1[k]; a.y = a1[k + 1];
      b.x = bp[k]; b.y = bp[k + 1];
      c = WMMA_F32(a, b, c);
    }
    const float* a2 = A2 + (size_t)row * K2s + 2 * kh;
    #pragma unroll 4
    for (int k = K1; k < Kk; k += 4) {
      v2f a, b;
      a.x = a2[k - K1]; a.y = a2[k - K1 + 1];
      b.x = bp[k];      b.y = bp[k + 1];
      c = WMMA_F32(a, b, c);
    }
  } else {
    const float* a1 = A1 + (size_t)row * K1 + 2 * kh;
    const float* bp = Bm + (size_t)(2 * kh) * Nn + col;
    #pragma unroll 4
    for (int k = 0; k < K1; k += 4) {
      v2f a, b;
      a.x = a1[k]; a.y = a1[k + 1];
      b.x = bp[(size_t)k * Nn];
      b.y = bp[(size_t)(k + 1) * Nn];
      c = WMMA_F32(a, b, c);
    }
    const float* a2 = A2 + (size_t)row * K2s + 2 * kh;
    #pragma unroll 4
    for (int k = K1; k < Kk; k += 4) {
      v2f a, b;
      a.x = a2[k - K1]; a.y = a2[k - K1 + 1];
      b.x = bp[(size_t)k * Nn];
      b.y = bp[(size_t)(k + 1) * Nn];
      c = WMMA_F32(a, b, c);
    }
  }
  float bval = bias ? bias[col] : 0.f;
  float* cp = Cm + ((size_t)mt * 16 + 8 * kh) * Nn + col;
  if (doTanh) {
    #pragma unroll
    for (int v = 0; v < 8; ++v) cp[(size_t)v * Nn] = tanhf(c[v] + bval);
  } else {
    #pragma unroll
    for (int v = 0; v < 8; ++v) cp[(size_t)v * Nn] = c[v] + bval;
  }
}

// ---------------- K5: mention pooling. scores = ment . qk[r] * 1/sqrt(D); softmax(M=4) ----------------
__global__ void k_pool(const float* __restrict__ seq, const float* __restrict__ qk,
                       const int* __restrict__ hts, const int* __restrict__ mpos,
                       int side, float* __restrict__ out) {
  int r = blockIdx.x;
  int b = r / Pc;
  int e = hts[r * 2 + side];
  int pos[Mc];
  #pragma unroll
  for (int m = 0; m < Mc; ++m) pos[m] = mpos[(b * Ec + e) * Mc + m] + 1;
  const float* qkr = qk + (size_t)r * Dc;
  const float* sb  = seq + (size_t)b * Lc * Dc;
  float acc[Mc] = {0.f, 0.f, 0.f, 0.f};
  for (int d = threadIdx.x; d < Dc; d += blockDim.x) {
    float q = qkr[d];
    #pragma unroll
    for (int m = 0; m < Mc; ++m) acc[m] += sb[(size_t)pos[m] * Dc + d] * q;
  }
  __shared__ float red[256];
  __shared__ float w[Mc];
  const float invs = 0.036084391824351615f;  // 1/sqrt(768)
  for (int m = 0; m < Mc; ++m) {
    red[threadIdx.x] = acc[m];
    __syncthreads();
    for (int s = 128; s > 0; s >>= 1) {
      if (threadIdx.x < s) red[threadIdx.x] += red[threadIdx.x + s];
      __syncthreads();
    }
    if (threadIdx.x == 0) w[m] = red[0] * invs;
    __syncthreads();
  }
  if (threadIdx.x == 0) {
    float mx = w[0];
    for (int m = 1; m < Mc; ++m) mx = fmaxf(mx, w[m]);
    float s = 0.f;
    for (int m = 0; m < Mc; ++m) { w[m] = expf(w[m] - mx); s += w[m]; }
    float inv = 1.f / s;
    for (int m = 0; m < Mc; ++m) w[m] *= inv;
  }
  __syncthreads();
  for (int d = threadIdx.x; d < Dc; d += blockDim.x) {
    float o = 0.f;
    #pragma unroll
    for (int m = 0; m < Mc; ++m) o += w[m] * sb[(size_t)pos[m] * Dc + d];
    out[(size_t)r * Dc + d] = o;
  }
}

// ---------------- K7: fused bilinear logits = (hs blk-outer ts) @ bil_W + bil_b ----------------
// A(16 x 49152) never materialized: a[m][kb*4096+i*64+j] = hs[m][kb*64+i] * ts[m][kb*64+j].
// Per kb-block each lane needs only 32 ts values -> preloaded to registers once per block.
// Partial N tile handled by clamping col (garbage cols computed, never stored).
__global__ void k_bil(const float* __restrict__ hs, const float* __restrict__ ts,
                      const float* __restrict__ bilW, const float* __restrict__ bilb,
                      float* __restrict__ out) {
  constexpr int MT = Rc / 16;            // 138 (exact)
  constexpr int NT = (NLc + 15) / 16;    // 7
  int wid = blockIdx.x * (blockDim.x >> 5) + (threadIdx.x >> 5);
  if (wid >= MT * NT) return;            // wave-uniform
  int mt = wid / NT, nt = wid - mt * NT;
  int lane = threadIdx.x & 31;
  int nl = lane & 15, kh = lane >> 4;
  int col = nt * 16 + nl;
  int colc = col < NLc ? col : (NLc - 1);   // clamp, no divergence in loop
  const float* hsrow = hs + (size_t)(mt * 16 + nl) * EMBc;
  const float* tsrow = ts + (size_t)(mt * 16 + nl) * EMBc;
  v8f c = {};
  #pragma unroll 1
  for (int kb = 0; kb < KBLK; ++kb) {
    const float* tsb = tsrow + kb * BLKc + 2 * kh;
    float tsv[32];
    #pragma unroll
    for (int q = 0; q < 16; ++q) {       // each lane's 32 ts values for this block
      tsv[2 * q]     = tsb[q * 4];
      tsv[2 * q + 1] = tsb[q * 4 + 1];
    }
    const float* hsb_ = hsrow + kb * BLKc;
    const float* bwkb = bilW + ((size_t)(kb * BLKc) * BLKc + 2 * kh) * NLc + colc;
    #pragma unroll 1
    for (int i = 0; i < BLKc; ++i) {
      float hv = hsb_[i];
      const float* bw = bwkb + (size_t)i * BLKc * NLc;
      #pragma unroll
      for (int q = 0; q < 16; ++q) {
        v2f a, b;
        a.x = hv * tsv[2 * q];
        a.y = hv * tsv[2 * q + 1];
        b.x = bw[(size_t)(q * 4) * NLc];
        b.y = bw[(size_t)(q * 4 + 1) * NLc];
        c = WMMA_F32(a, b, c);
      }
    }
  }
  float* op = out + ((size_t)mt * 16 + 8 * kh) * NLc + col;
  if (nt != NT - 1) {                   // wave-uniform: all 16 cols valid
    #pragma unroll
    for (int v = 0; v < 8; ++v) op[(size_t)v * NLc] = c[v] + bilb[col];
  } else {                              // last N tile: cols 96..111, NL=97
    #pragma unroll
    for (int v = 0; v < 8; ++v) {
      if (col < NLc) op[(size_t)v * NLc] = c[v] + bilb[col];
    }
  }
}

// ---------------- launcher ----------------
extern "C" void kernel_launch(void* const* d_in, const int* in_sizes, int n_in,
                              void* d_out, int out_size, void* d_ws, size_t ws_size,
                              hipStream_t stream) {
  (void)in_sizes; (void)n_in; (void)out_size; (void)ws_size;
  const float* seq   = (const float*)d_in[0];
  const float* att   = (const float*)d_in[1];
  const float* Wq    = (const float*)d_in[2];
  const float* Wk    = (const float*)d_in[3];
  const float* headW = (const float*)d_in[4];
  const float* headb = (const float*)d_in[5];
  const float* tailW = (const float*)d_in[6];
  const float* tailb = (const float*)d_in[7];
  const float* bilW  = (const float*)d_in[8];
  const float* bilb  = (const float*)d_in[9];
  const int*   mpos  = (const int*)d_in[10];
  const int*   hts   = (const int*)d_in[11];
  float* ws  = (float*)d_ws;
  float* ent = ws + OFF_ENT;
  float* ht  = ws + OFF_HT;
  float* rs  = ws + OFF_RS;
  float* qc  = ws + OFF_QC;
  float* qk  = ws + OFF_QK;
  float* he  = ws + OFF_HE;
  float* te  = ws + OFF_TE;
  float* hsb = ws + OFF_HS;
  float* tsb = ws + OFF_TS;

  k_ent<<<Bc * Ec * Hc, 256, 0, stream>>>(att, mpos, ent);
  k_ht<<<Rc, 256, 0, stream>>>(ent, hts, ht);
  {
    int waves = Bc * ((Pc + 15) / 16) * (Dc / 16);            // 4*35*48
    k_rs<<<(waves + 7) / 8, 256, 0, stream>>>(ht, seq, rs);
  }
  {
    int waves = (Rc / 16) * (Dc / 16);                        // 138*48
    // q_c = rs @ W_q
    k_gemm2<<<(waves + 7) / 8, 256, 0, stream>>>(rs, rs, Dc, Wq, 0, nullptr, 0, qc, Rc, Dc, Dc);
    // qk = q_c @ W_k^T   (folds W_k into the query: scores = ment . qk)
    k_gemm2<<<(waves + 7) / 8, 256, 0, stream>>>(qc, qc, Dc, Wk, 1, nullptr, 0, qk, Rc, Dc, Dc);
  }
  k_pool<<<Rc, 256, 0, stream>>>(seq, qk, hts, mpos, 0, he);
  k_pool<<<Rc, 256, 0, stream>>>(seq, qk, hts, mpos, 1, te);
  {
    int waves = (Rc / 16) * (EMBc / 16);                      // 138*48
    // hs = tanh([h_ent | rs] @ head_W + head_b), ts likewise
    k_gemm2<<<(waves + 7) / 8, 256, 0, stream>>>(he, rs, Dc, headW, 0, headb, 1, hsb, Rc, 2 * Dc, EMBc);
    k_gemm2<<<(waves + 7) / 8, 256, 0, stream>>>(te, rs, Dc, tailW, 0, tailb, 1, tsb, Rc, 2 * Dc, EMBc);
  }
  {
    int waves = (Rc / 16) * ((NLc + 15) / 16);                // 138*7
    k_bil<<<(waves + 7) / 8, 256, 0, stream>>>(hsb, tsb, bilW, bilb, (float*)d_out);
  }
}